// FullyConnectedWithTripletLoss_11914239279333
// MI455X (gfx1250) — compile-verified
//
#include <hip/hip_runtime.h>
#include <hip/hip_bf16.h>

typedef __attribute__((ext_vector_type(2))) float v2f;
typedef __attribute__((ext_vector_type(8))) float v8f;

#define N_PTS 1024
#define F_IN  512
#define F_OUT 128

__device__ __forceinline__ float neg_inf() { return -__builtin_inff(); }
__device__ __forceinline__ float pos_inf() { return  __builtin_inff(); }

// hp starts at -inf, all candidate values are >= 0 distances: signed-int compare
// on float bit patterns preserves ordering here (nonneg floats are nonneg ints,
// -inf bits are a very negative int). Same argument for hn with +inf init.
__global__ void k_init(float* __restrict__ hp, float* __restrict__ hn) {
    int i = blockIdx.x * blockDim.x + threadIdx.x;
    if (i < N_PTS) { hp[i] = neg_inf(); hn[i] = pos_inf(); }
}

// h = inp @ W + b  via V_WMMA_F32_16X16X4_F32. One wave per 16x16 tile of h.
__global__ void k_linear(const float* __restrict__ inp, const float* __restrict__ W,
                         const float* __restrict__ b, float* __restrict__ h) {
    const int lane = threadIdx.x & 31;
    const int wid  = threadIdx.x >> 5;
    const int tile = blockIdx.x * 8 + wid;      // 512 tiles total
    const int ti = tile >> 3;                   // 0..63 (rows of h)
    const int tj = tile & 7;                    // 0..7  (cols of h)
    const int i0 = ti * 16, j0 = tj * 16;
    const int m    = lane & 15;                 // row (A) / col (B) within tile
    const int koff = (lane >> 4) * 2;           // K sub-offset per half-wave

    const float* arow = inp + (size_t)(i0 + m) * F_IN + koff;

    v8f c = {};
    #pragma unroll 8
    for (int k = 0; k < F_IN; k += 4) {
        v2f a = *(const v2f*)(arow + k);
        v2f bb;
        bb.x = W[(size_t)(k + koff)     * F_OUT + j0 + m];
        bb.y = W[(size_t)(k + koff + 1) * F_OUT + j0 + m];
        c = __builtin_amdgcn_wmma_f32_16x16x4_f32(false, a, false, bb,
                                                  (short)0, c, false, false);
    }
    const float bias = b[j0 + m];
    const int mbase = (lane >> 4) * 8;          // C/D: vgpr r -> row r (+8 for hi half)
    #pragma unroll
    for (int r = 0; r < 8; ++r)
        h[(size_t)(i0 + mbase + r) * F_OUT + j0 + m] = c[r] + bias;
}

// nrm[i] = ||h_i||^2 ; one wave per row (32 lanes x float4 = 128 floats)
__global__ void k_norms(const float* __restrict__ h, float* __restrict__ nrm) {
    const int lane = threadIdx.x & 31;
    const int wid  = threadIdx.x >> 5;
    const int row  = blockIdx.x * 8 + wid;
    const float4* p = (const float4*)(h + (size_t)row * F_OUT);
    float4 v = p[lane];
    float s = v.x * v.x + v.y * v.y + v.z * v.z + v.w * v.w;
    #pragma unroll
    for (int off = 1; off < 32; off <<= 1) s += __shfl_xor(s, off, 32);
    if (lane == 0) nrm[row] = s;
}

// Gram tile via WMMA, convert to distances, masked row max/min, atomic combine.
__global__ void k_pair(const float* __restrict__ h, const float* __restrict__ nrm,
                       const int* __restrict__ targets,
                       float* __restrict__ hp, float* __restrict__ hn) {
    const int lane = threadIdx.x & 31;
    const int wid  = threadIdx.x >> 5;
    const int tile = blockIdx.x * 8 + wid;      // 4096 tiles (64x64)
    const int ti = tile >> 6;
    const int tj = tile & 63;
    const int i0 = ti * 16, j0 = tj * 16;
    const int m    = lane & 15;
    const int koff = (lane >> 4) * 2;

    const float* arow = h + (size_t)(i0 + m) * F_OUT + koff;  // A = h rows i0..i0+15
    const float* brow = h + (size_t)(j0 + m) * F_OUT + koff;  // B = h^T cols j0..j0+15

    v8f c = {};
    #pragma unroll 8
    for (int k = 0; k < F_OUT; k += 4) {
        v2f a  = *(const v2f*)(arow + k);
        v2f bb = *(const v2f*)(brow + k);
        c = __builtin_amdgcn_wmma_f32_16x16x4_f32(false, a, false, bb,
                                                  (short)0, c, false, false);
    }

    const int hs   = lane >> 4;
    const int jcol = j0 + m;
    const float nj = nrm[jcol];
    const int tj_cls = targets[jcol];

    #pragma unroll
    for (int r = 0; r < 8; ++r) {
        const int irow = i0 + hs * 8 + r;
        const float d2 = nrm[irow] + nj - 2.0f * c[r];
        const float d  = sqrtf(fmaxf(d2, 0.0f));
        const bool same = (targets[irow] == tj_cls);
        float pv = (same && (irow != jcol)) ? d : neg_inf();
        float nv = same ? pos_inf() : d;
        // reduce across the 16 lanes holding this row's 16 columns
        #pragma unroll
        for (int off = 1; off < 16; off <<= 1) {
            pv = fmaxf(pv, __shfl_xor(pv, off, 16));
            nv = fminf(nv, __shfl_xor(nv, off, 16));
        }
        if (m == 0) {
            if (pv > neg_inf()) atomicMax((int*)&hp[irow], __float_as_int(pv));
            if (nv < pos_inf()) atomicMin((int*)&hn[irow], __float_as_int(nv));
        }
    }
}

// loss = sum_i softplus(hp[i] - hn[i]); single 1024-thread block.
__global__ void k_final(const float* __restrict__ hp, const float* __restrict__ hn,
                        float* __restrict__ out) {
    __shared__ float sdata[32];
    const int i = threadIdx.x;
    const float x = hp[i] - hn[i];
    // stable softplus: max(x,0) + log1p(exp(-|x|)); x=-inf -> 0 (empty pos set)
    float sp = fmaxf(x, 0.0f) + log1pf(expf(-fabsf(x)));
    const int lane = i & 31, wid = i >> 5;
    #pragma unroll
    for (int off = 1; off < 32; off <<= 1) sp += __shfl_xor(sp, off, 32);
    if (lane == 0) sdata[wid] = sp;
    __syncthreads();
    if (wid == 0) {
        float v = sdata[lane];
        #pragma unroll
        for (int off = 1; off < 32; off <<= 1) v += __shfl_xor(v, off, 32);
        if (lane == 0) out[0] = v;
    }
}

extern "C" void kernel_launch(void* const* d_in, const int* in_sizes, int n_in,
                              void* d_out, int out_size, void* d_ws, size_t ws_size,
                              hipStream_t stream) {
    const float* inp     = (const float*)d_in[0];   // [1024,512]
    const int*   targets = (const int*)  d_in[1];   // [1024]
    const float* W       = (const float*)d_in[2];   // [512,128]
    const float* b       = (const float*)d_in[3];   // [128]
    float* out = (float*)d_out;

    char* ws = (char*)d_ws;
    float* h   = (float*)ws;                                   // 1024*128 floats
    float* nrm = (float*)(ws + (size_t)N_PTS * F_OUT * 4);     // 1024 floats
    float* hp  = nrm + N_PTS;                                  // 1024 floats
    float* hn  = hp + N_PTS;                                   // 1024 floats

    k_init  <<<4,   256, 0, stream>>>(hp, hn);
    k_linear<<<64,  256, 0, stream>>>(inp, W, b, h);           // 512 tiles / 8 waves
    k_norms <<<128, 256, 0, stream>>>(h, nrm);                 // 1024 rows / 8 waves
    k_pair  <<<512, 256, 0, stream>>>(h, nrm, targets, hp, hn);// 4096 tiles / 8 waves
    k_final <<<1,  1024, 0, stream>>>(hp, hn, out);
}